// RNN_Model_5935644803816
// MI455X (gfx1250) — compile-verified
//
#include <hip/hip_runtime.h>
#include <hip/hip_bf16.h>

// ---------------------------------------------------------------------------
// Problem: T=32768 sequential LSTM steps over 188 independent rows, G=4.
//   Phase 1 (serial, latency-bound): one lane per row, state in VGPRs,
//            per-step 32-lane butterfly reduce of ot*Ww -> 6 partials/step.
//   Phase 2 (parallel, WMMA): out[t,g] = bw + sum_b partial[t,b,g] done as
//            D = A x ones-column via v_wmma_f32_16x16x4_f32.
// Scratch use: 32768 * 6 * 4 floats = 3 MB in d_ws.
// ---------------------------------------------------------------------------

#define T_STEPS 32768
#define NROWS   188

typedef __attribute__((ext_vector_type(2))) float v2f;
typedef __attribute__((ext_vector_type(8))) float v8f;

__device__ __forceinline__ float fast_tanh(float x) {
#if __has_builtin(__builtin_amdgcn_tanhf)
    return __builtin_amdgcn_tanhf(x);          // gfx1250 v_tanh_f32 (trans op)
#else
    // tanh(x) = 2/(1+exp(-2x)) - 1 ; exp via v_exp_f32 (2^y), rcp via v_rcp_f32
    float e = __builtin_amdgcn_exp2f(x * -2.8853900817779268f); // -2*log2(e)
    return __builtin_amdgcn_rcpf(1.0f + e) * 2.0f - 1.0f;
#endif
}
__device__ __forceinline__ float fast_sigmoid(float x) {
    return 0.5f + 0.5f * fast_tanh(0.5f * x);
}

// ---------------------------------------------------------------------------
// Phase 1: the serial recurrence. 6 blocks x 32 lanes = 192 lanes, 188 active
// rows (tail lanes clamp their loads and contribute p=0 via Ww=0, keeping
// EXEC fully uniform). No cross-lane traffic on the carry path; the butterfly
// reduction only feeds the stored output partials.
// ---------------------------------------------------------------------------
__global__ __launch_bounds__(32, 1)
void lstm_recurrence(const float* __restrict__ input,
                     const float* __restrict__ h0,
                     const float* __restrict__ c0,
                     const float* __restrict__ Wf, const float* __restrict__ bf,
                     const float* __restrict__ Wi, const float* __restrict__ bi,
                     const float* __restrict__ Wc, const float* __restrict__ bc,
                     const float* __restrict__ Wo, const float* __restrict__ bo,
                     const float* __restrict__ Ww,
                     float* __restrict__ partials /* [T][6][4] */) {
    const int lane = threadIdx.x;
    const int blk  = blockIdx.x;
    const int row  = blk * 32 + lane;
    const int nidx = (row < NROWS) ? row : (NROWS - 1);   // clamp OOB lanes
    const float wwn = (row < NROWS) ? Ww[row] : 0.0f;     // zero their output

    // Hoist all (uniform) weights into registers: 16 rows of [x,h0..h3] + bias.
    float W[16][5], B[16];
    {
        const float* wp[4] = {Wf, Wi, Wc, Wo};
        const float* bp[4] = {bf, bi, bc, bo};
#pragma unroll
        for (int mtx = 0; mtx < 4; ++mtx) {
#pragma unroll
            for (int g = 0; g < 4; ++g) {
                B[mtx * 4 + g] = bp[mtx][g];
#pragma unroll
                for (int j = 0; j < 5; ++j)
                    W[mtx * 4 + g][j] = wp[mtx][g * 5 + j];
            }
        }
    }

    float h[4], c[4];
#pragma unroll
    for (int g = 0; g < 4; ++g) {
        h[g] = h0[nidx * 4 + g];
        c[g] = c0[nidx * 4 + g];
    }

    const float* xp   = input + nidx;          // advance by NROWS per step
    float*       outp = partials + blk * 4;    // advance by 24 per step

    for (int t = 0; t < T_STEPS; ++t) {
        const float x = *xp;
        xp += NROWS;

        // 16 independent length-5 dot products (good ILP for a lone wave).
        float acc[16];
#pragma unroll
        for (int m = 0; m < 16; ++m) {
            acc[m] = fmaf(W[m][0], x,
                     fmaf(W[m][1], h[0],
                     fmaf(W[m][2], h[1],
                     fmaf(W[m][3], h[2],
                     fmaf(W[m][4], h[3], B[m])))));
        }

        float p[4];
#pragma unroll
        for (int g = 0; g < 4; ++g) {
            const float ft = fast_sigmoid(acc[g]);
            const float it = fast_sigmoid(acc[4 + g]);
            const float cb = fast_tanh   (acc[8 + g]);
            const float ot = fast_sigmoid(acc[12 + g]);
            const float cn = fmaf(ft, c[g], it * cb);
            c[g] = cn;
            h[g] = fast_tanh(cn) * ot;
            p[g] = ot * wwn;                   // output-projection product
        }

        // 32-lane butterfly reduction (off the carry critical path).
#pragma unroll
        for (int g = 0; g < 4; ++g) {
#pragma unroll
            for (int off = 16; off > 0; off >>= 1)
                p[g] += __shfl_xor(p[g], off, 32);
        }
        if (lane == 0) {
            *(float4*)outp = make_float4(p[0], p[1], p[2], p[3]);
        }
        outp += 24;
    }
}

// ---------------------------------------------------------------------------
// Phase 2: out[o] = bw + sum_{b=0..5} partial[t,b,g],  o = t*4+g.
// Each wave produces 16 outputs via two V_WMMA_F32_16X16X4_F32 ops:
//   A1[m][k] = partial(b=k),  A2[m][0..3] = {partial(4), partial(5), bw, 0}
//   B ones in column N=0  =>  D[m][0] = row sums.
// A layout (16x4 f32): lane l<16 holds (A[l][0],A[l][1]); lane l+16 holds
// (A[l][2],A[l][3]). B: column N = lane%16, so lanes 0 and 16 = (1,1) puts a
// 1 in every K-row of column 0 regardless of row striping. D[m][0] lands in
// lane 0 (m=0..7) and lane 16 (m=8..15), one per accumulator VGPR.
// ---------------------------------------------------------------------------
__global__ __launch_bounds__(128, 1)
void out_reduce_wmma(const float* __restrict__ partials,
                     const float* __restrict__ bwp,
                     float* __restrict__ out) {
    const int lane  = threadIdx.x & 31;
    const int wave  = (blockIdx.x * blockDim.x + threadIdx.x) >> 5;
    const int obase = wave * 16;               // 131072 outputs / 16 per wave
    const int m     = lane & 15;
    const int o     = obase + m;
    const int t     = o >> 2;
    const int g     = o & 3;
    const float bw  = *bwp;

    const float* P  = partials + t * 24 + g;   // + b*4 selects partial b
    const bool   hi = lane >= 16;

    // Branchless operand builds: EXEC stays all-1s into the WMMAs.
    v2f a1, a2, b;
    a1.x = hi ? P[2 * 4] : P[0 * 4];
    a1.y = hi ? P[3 * 4] : P[1 * 4];
    a2.x = hi ? bw       : P[4 * 4];
    a2.y = hi ? 0.0f     : P[5 * 4];
    const float one = (m == 0) ? 1.0f : 0.0f;
    b.x = one;
    b.y = one;

    v8f acc = {};
    acc = __builtin_amdgcn_wmma_f32_16x16x4_f32(false, a1, false, b,
                                                (short)0, acc, false, false);
    acc = __builtin_amdgcn_wmma_f32_16x16x4_f32(false, a2, false, b,
                                                (short)0, acc, false, false);

    if (m == 0) {                              // lanes 0 and 16 hold column 0
        const int base = obase + (hi ? 8 : 0);
        *(float4*)(out + base)     = make_float4(acc[0], acc[1], acc[2], acc[3]);
        *(float4*)(out + base + 4) = make_float4(acc[4], acc[5], acc[6], acc[7]);
    }
}

// ---------------------------------------------------------------------------
extern "C" void kernel_launch(void* const* d_in, const int* in_sizes, int n_in,
                              void* d_out, int out_size, void* d_ws, size_t ws_size,
                              hipStream_t stream) {
    const float* input = (const float*)d_in[0];
    const float* h0    = (const float*)d_in[1];
    const float* c0    = (const float*)d_in[2];
    const float* Wf    = (const float*)d_in[3];
    const float* bf    = (const float*)d_in[4];
    const float* Wi    = (const float*)d_in[5];
    const float* bi    = (const float*)d_in[6];
    const float* Wc    = (const float*)d_in[7];
    const float* bc    = (const float*)d_in[8];
    const float* Wo    = (const float*)d_in[9];
    const float* bo    = (const float*)d_in[10];
    const float* Ww    = (const float*)d_in[11];
    const float* bwp   = (const float*)d_in[12];
    float*       out   = (float*)d_out;
    float*       parts = (float*)d_ws;         // needs 32768*24*4 = 3 MB

    lstm_recurrence<<<6, 32, 0, stream>>>(input, h0, c0, Wf, bf, Wi, bi,
                                          Wc, bc, Wo, bo, Ww, parts);

    // 32768*4 = 131072 outputs; 16 per wave -> 8192 waves -> 2048 blocks of 128.
    out_reduce_wmma<<<2048, 128, 0, stream>>>(parts, bwp, out);
}